// Criterion_89180700934218
// MI455X (gfx1250) — compile-verified
//
#include <hip/hip_runtime.h>

// CRAFT-style OHEM loss on gfx1250.
// Strategy: two-level float-bit-pattern histogram select (radix select) for the
// top-k negative losses, fused with the MSE computation. Streaming passes are
// HBM-bound (~302 MB/pass @ 23.3 TB/s); histograms live in LDS.
// CDNA5 paths: v_wmma_f32_16x16x4_f32 used as the wave reduction primitive
// (B = ones => row sums), global_prefetch_b8 on the streaming loads.

#define BATCH   32
#define NPIX    (768 * 768)
#define NPROB   (BATCH * 2)
#define CBINS   1024          // coarse: bits >> 20 (loss in [0,1] -> 0..1016)
#define FBINS   4096          // fine:  (bits >> 8) & 0xFFF within coarse bin
#define POS_THR 0.1f
#define TOPK_NO_POS 500
#define NEG_RATIO   3

typedef __attribute__((ext_vector_type(2))) float v2f;
typedef __attribute__((ext_vector_type(8))) float v8f;

// Wave32 all-lane sum via V_WMMA_F32_16X16X4_F32 with B = ones.
// A layout (16x4 f32): vgpr0 lanes0-15 -> A[m][0], lanes16-31 -> A[m][2];
// vgpr1 -> A[m][1]/A[m][3] (set to 0). D[m][n] = v[m] + v[m+16], replicated
// across columns. Sum of the 8 D vgprs gives rows 0..7 (lanes<16) / 8..15
// (lanes>=16); one shfl_xor(16) completes the 32-lane sum.
// Requires EXEC == all ones (call only from uniform control flow).
__device__ __forceinline__ float wave_sum_wmma(float v) {
  v2f a; a[0] = v;   a[1] = 0.0f;
  v2f b; b[0] = 1.0f; b[1] = 1.0f;
  v8f c = {};
  c = __builtin_amdgcn_wmma_f32_16x16x4_f32(false, a, false, b,
                                            (short)0, c, false, false);
  float s = c[0] + c[1] + c[2] + c[3] + c[4] + c[5] + c[6] + c[7];
  s += __shfl_xor(s, 16, 32);
  return s;
}

// ---------------------------------------------------------------- pass 1 ----
// Fused MSE + positive stats + coarse histogram (count & sum per bin).
// One block row per sample b (blockIdx.y); handles both channels so pred's
// interleaved float2 cachelines are read exactly once.
__global__ void pass1_hist(const float2* __restrict__ pred,
                           const float*  __restrict__ region,
                           const float*  __restrict__ affin,
                           unsigned* __restrict__ g_cnt,   // [NPROB][CBINS]
                           float*    __restrict__ g_sum,   // [NPROB][CBINS]
                           float*    __restrict__ g_pcnt,  // [NPROB]
                           float*    __restrict__ g_psum)  // [NPROB]
{
  __shared__ unsigned scnt[2][CBINS];
  __shared__ float    ssum[2][CBINS];
  const int b   = blockIdx.y;
  const int tid = threadIdx.x;

  for (int i = tid; i < 2 * CBINS; i += blockDim.x) {
    (&scnt[0][0])[i] = 0u;
    (&ssum[0][0])[i] = 0.0f;
  }
  __syncthreads();

  float pc0 = 0.f, ps0 = 0.f, pc1 = 0.f, ps1 = 0.f;
  const long base   = (long)b * NPIX;
  const int  stride = gridDim.x * blockDim.x;

  for (int i = blockIdx.x * blockDim.x + tid; i < NPIX; i += stride) {
    float2 pv = pred[base + i];
    float  rl = region[base + i];
    float  al = affin[base + i];
    __builtin_prefetch(&pred[base + i + stride],   0, 0);  // global_prefetch_b8
    __builtin_prefetch(&region[base + i + stride], 0, 0);
    __builtin_prefetch(&affin[base + i + stride],  0, 0);

    float d0 = pv.x - rl, l0 = d0 * d0;
    float d1 = pv.y - al, l1 = d1 * d1;

    if (rl >= POS_THR) { pc0 += 1.0f; ps0 += l0; }
    else {
      unsigned bin = __float_as_uint(l0) >> 20;   // 0..1016
      atomicAdd(&scnt[0][bin], 1u);
      atomicAdd(&ssum[0][bin], l0);
    }
    if (al >= POS_THR) { pc1 += 1.0f; ps1 += l1; }
    else {
      unsigned bin = __float_as_uint(l1) >> 20;
      atomicAdd(&scnt[1][bin], 1u);
      atomicAdd(&ssum[1][bin], l1);
    }
  }

  // Wave-level reduction on the matrix pipe (EXEC all-ones here).
  float wc0 = wave_sum_wmma(pc0);
  float ws0 = wave_sum_wmma(ps0);
  float wc1 = wave_sum_wmma(pc1);
  float ws1 = wave_sum_wmma(ps1);
  if ((tid & 31) == 0) {
    atomicAdd(&g_pcnt[2 * b + 0], wc0);
    atomicAdd(&g_psum[2 * b + 0], ws0);
    atomicAdd(&g_pcnt[2 * b + 1], wc1);
    atomicAdd(&g_psum[2 * b + 1], ws1);
  }

  __syncthreads();
  for (int i = tid; i < CBINS; i += blockDim.x) {
    if (scnt[0][i])         atomicAdd(&g_cnt[(2 * b + 0) * CBINS + i], scnt[0][i]);
    if (ssum[0][i] != 0.0f) atomicAdd(&g_sum[(2 * b + 0) * CBINS + i], ssum[0][i]);
    if (scnt[1][i])         atomicAdd(&g_cnt[(2 * b + 1) * CBINS + i], scnt[1][i]);
    if (ssum[1][i] != 0.0f) atomicAdd(&g_sum[(2 * b + 1) * CBINS + i], ssum[1][i]);
  }
}

// ------------------------------------------------------- coarse selection ---
// Per problem: k, pivot coarse bin t, count & sum strictly above t.
__global__ void select_coarse(const unsigned* __restrict__ g_cnt,
                              const float*    __restrict__ g_sum,
                              const float*    __restrict__ g_pcnt,
                              int* __restrict__ o_t, int* __restrict__ o_k,
                              int* __restrict__ o_c, float* __restrict__ o_s)
{
  int pr = blockIdx.x * blockDim.x + threadIdx.x;
  if (pr >= NPROB) return;

  int  p  = (int)(g_pcnt[pr] + 0.5f);
  long nn = (long)NPIX - p;
  long k  = (p > 0) ? ((nn < (long)NEG_RATIO * p) ? nn : (long)NEG_RATIO * p)
                    : (long)TOPK_NO_POS;

  int cA = 0; float sA = 0.0f; int t = -1;
  if (k > 0) {
    // k <= nn == total negative count, so the walk always terminates with a bin.
    for (t = CBINS - 1; t > 0; --t) {
      unsigned c = g_cnt[pr * CBINS + t];
      if ((long)cA + c >= k) break;
      cA += (int)c;
      sA += g_sum[pr * CBINS + t];
    }
  }
  o_t[pr] = (k > 0) ? t : -1;
  o_k[pr] = (int)k;
  o_c[pr] = cA;
  o_s[pr] = sA;
}

// ---------------------------------------------------------------- pass 2 ----
// Re-stream, refine inside the pivot coarse bin: 4096-bin fine histogram.
// One (sample, channel) per block row/plane.
__global__ void pass2_fine(const float2* __restrict__ pred,
                           const float*  __restrict__ region,
                           const float*  __restrict__ affin,
                           const int*    __restrict__ o_t,
                           unsigned* __restrict__ f_cnt,   // [NPROB][FBINS]
                           float*    __restrict__ f_sum)   // [NPROB][FBINS]
{
  __shared__ unsigned fc[FBINS];
  __shared__ float    fs[FBINS];
  const int b  = blockIdx.y;
  const int ch = blockIdx.z;
  const int pr = 2 * b + ch;
  const int t  = o_t[pr];
  const int tid = threadIdx.x;

  for (int i = tid; i < FBINS; i += blockDim.x) { fc[i] = 0u; fs[i] = 0.0f; }
  __syncthreads();

  if (t >= 0) {
    const long  base   = (long)b * NPIX;
    const int   stride = gridDim.x * blockDim.x;
    const float* lab   = ch ? affin : region;
    for (int i = blockIdx.x * blockDim.x + tid; i < NPIX; i += stride) {
      float2 pv = pred[base + i];
      float  lb = lab[base + i];
      __builtin_prefetch(&pred[base + i + stride], 0, 0);
      __builtin_prefetch(&lab[base + i + stride],  0, 0);
      float pc = ch ? pv.y : pv.x;
      float d  = pc - lb, ls = d * d;
      unsigned bits = __float_as_uint(ls);
      if (lb < POS_THR && (int)(bits >> 20) == t) {
        unsigned f = (bits >> 8) & (FBINS - 1);
        atomicAdd(&fc[f], 1u);
        atomicAdd(&fs[f], ls);
      }
    }
  }

  __syncthreads();
  for (int i = tid; i < FBINS; i += blockDim.x) {
    if (fc[i])         atomicAdd(&f_cnt[pr * FBINS + i], fc[i]);
    if (fs[i] != 0.0f) atomicAdd(&f_sum[pr * FBINS + i], fs[i]);
  }
}

// -------------------------------------------------------------- finalize ----
__global__ void finalize(const unsigned* __restrict__ f_cnt,
                         const float*    __restrict__ f_sum,
                         const int* __restrict__ o_t, const int* __restrict__ o_k,
                         const int* __restrict__ o_c, const float* __restrict__ o_s,
                         const float* __restrict__ g_pcnt,
                         const float* __restrict__ g_psum,
                         float* __restrict__ out)
{
  __shared__ float acc[64];
  const int pr = threadIdx.x;
  float contrib = 0.0f;

  if (pr < NPROB) {
    float p    = g_pcnt[pr];
    float posi = (p > 0.0f) ? g_psum[pr] / p : 0.0f;
    float nega = 0.0f;
    int   k    = o_k[pr];
    int   t    = o_t[pr];
    if (k > 0 && t >= 0) {
      int   r0 = k - o_c[pr];        // values still needed from coarse bin t
      int   c2 = 0; float s2 = 0.0f; int f;
      for (f = FBINS - 1; f > 0; --f) {
        unsigned c = f_cnt[pr * FBINS + f];
        if (c2 + (int)c >= r0) break;
        c2 += (int)c;
        s2 += f_sum[pr * FBINS + f];
      }
      int      r  = r0 - c2;         // 1 <= r <= fc[f]
      unsigned cf = f_cnt[pr * FBINS + f];
      // fine-bin values share 24 leading bits -> mean approximation error < 2^-15
      float partial = (cf > 0u) ? (float)r * (f_sum[pr * FBINS + f] / (float)cf) : 0.0f;
      nega = (o_s[pr] + s2 + partial) / (float)k;
    }
    contrib = posi + nega;
  }

  acc[threadIdx.x] = contrib;
  __syncthreads();
  for (int off = 32; off > 0; off >>= 1) {
    if (threadIdx.x < off) acc[threadIdx.x] += acc[threadIdx.x + off];
    __syncthreads();
  }
  if (threadIdx.x == 0) out[0] = acc[0] / (float)BATCH;  // char/B + affi/B
}

// ----------------------------------------------------------------------------
extern "C" void kernel_launch(void* const* d_in, const int* in_sizes, int n_in,
                              void* d_out, int out_size, void* d_ws, size_t ws_size,
                              hipStream_t stream) {
  (void)in_sizes; (void)n_in; (void)out_size; (void)ws_size;
  const float2* pred   = (const float2*)d_in[0];
  const float*  region = (const float*)d_in[1];
  const float*  affin  = (const float*)d_in[2];
  float*        out    = (float*)d_out;

  char* ws = (char*)d_ws;
  constexpr size_t OFF_GCNT = 0;                                    // 256 KB
  constexpr size_t OFF_GSUM = OFF_GCNT + (size_t)NPROB * CBINS * 4; // 256 KB
  constexpr size_t OFF_PCNT = OFF_GSUM + (size_t)NPROB * CBINS * 4;
  constexpr size_t OFF_PSUM = OFF_PCNT + 1024;
  constexpr size_t OFF_T    = OFF_PSUM + 1024;
  constexpr size_t OFF_K    = OFF_T + 1024;
  constexpr size_t OFF_C    = OFF_K + 1024;
  constexpr size_t OFF_S    = OFF_C + 1024;
  constexpr size_t OFF_FCNT = OFF_S + 1024;                         // 1 MB
  constexpr size_t OFF_FSUM = OFF_FCNT + (size_t)NPROB * FBINS * 4; // 1 MB
  constexpr size_t WS_TOTAL = OFF_FSUM + (size_t)NPROB * FBINS * 4; // ~2.6 MB

  unsigned* g_cnt  = (unsigned*)(ws + OFF_GCNT);
  float*    g_sum  = (float*)   (ws + OFF_GSUM);
  float*    g_pcnt = (float*)   (ws + OFF_PCNT);
  float*    g_psum = (float*)   (ws + OFF_PSUM);
  int*      o_t    = (int*)     (ws + OFF_T);
  int*      o_k    = (int*)     (ws + OFF_K);
  int*      o_c    = (int*)     (ws + OFF_C);
  float*    o_s    = (float*)   (ws + OFF_S);
  unsigned* f_cnt  = (unsigned*)(ws + OFF_FCNT);
  float*    f_sum  = (float*)   (ws + OFF_FSUM);

  hipMemsetAsync(d_ws, 0, WS_TOTAL, stream);

  pass1_hist<<<dim3(64, BATCH), 256, 0, stream>>>(
      pred, region, affin, g_cnt, g_sum, g_pcnt, g_psum);

  select_coarse<<<1, 64, 0, stream>>>(g_cnt, g_sum, g_pcnt, o_t, o_k, o_c, o_s);

  pass2_fine<<<dim3(32, BATCH, 2), 256, 0, stream>>>(
      pred, region, affin, o_t, f_cnt, f_sum);

  finalize<<<1, 64, 0, stream>>>(f_cnt, f_sum, o_t, o_k, o_c, o_s,
                                 g_pcnt, g_psum, out);
}